// RTDetrHungarianMatcher_62508954026120
// MI455X (gfx1250) — compile-verified
//
#include <hip/hip_runtime.h>

#define ALPHA_F   0.25f
#define CLASS_W   2.0f
#define BBOX_W    5.0f
#define GIOU_W    2.0f
#define FOCAL_EPS 1e-8f

#define NCLASS 80
#define N_TILE 32
#define T_TILE 256

__device__ __forceinline__ float fast_rcp(float x) { return __builtin_amdgcn_rcpf(x); }

// ---------------------------------------------------------------------------
// Kernel 1: precompute scaled focal class cost cc[n][c] = CLASS_W*(pos - neg).
// Turns 25.6M per-pair transcendental evaluations into 640K (then gathered).
// ---------------------------------------------------------------------------
__global__ __launch_bounds__(256)
void cc_precompute_kernel(const float* __restrict__ logits,
                          float* __restrict__ cc, int total)
{
    int i = blockIdx.x * 256 + threadIdx.x;
    if (i >= total) return;
    float x   = logits[i];
    float p   = fast_rcp(1.0f + __expf(-x));     // sigmoid
    float omp = 1.0f - p;
    float neg = (1.0f - ALPHA_F) * p * p   * (-__logf(omp + FOCAL_EPS));
    float pos = ALPHA_F          * omp*omp * (-__logf(p   + FOCAL_EPS));
    cc[i] = CLASS_W * (pos - neg);
}

// ---------------------------------------------------------------------------
// Kernel 2: pairwise cost. lane <-> t (coalesced stores), 32 n-rows per block.
// cc tile staged global->LDS via CDNA5 async-to-LDS ops (ASYNCcnt).
// Output written with non-temporal stores (write-once 102 MB stream).
// ---------------------------------------------------------------------------
__global__ __launch_bounds__(256)
void cost_kernel(const float* __restrict__ pred_boxes,
                 const int*  __restrict__ labels,
                 const float* __restrict__ tboxes,
                 const float* __restrict__ cc,
                 float* __restrict__ out,
                 int Nq, int Nt)
{
    __shared__ __align__(16) float  sh_cc[N_TILE * NCLASS];  // 10240 B
    __shared__ __align__(16) float4 sh_pm[N_TILE * 3];       // corners / cbox / area

    const int tid = threadIdx.x;
    const int t   = blockIdx.x * T_TILE + tid;
    const int n0  = blockIdx.y * N_TILE;

    // ---- async stage of cc tile: N_TILE*NCLASS contiguous floats ----
    {
        const float* ccbase = cc + (size_t)n0 * NCLASS;      // uniform -> SGPR pair
        unsigned lds0 = (unsigned)(size_t)(&sh_cc[0]);       // LDS byte offset
        for (int idx = tid; idx < (N_TILE * NCLASS) / 4; idx += 256) {
            unsigned boff = (unsigned)idx * 16u;
            asm volatile("global_load_async_to_lds_b128 %0, %1, %2"
                         :: "v"(lds0 + boff), "v"(boff), "s"(ccbase)
                         : "memory");
        }
    }

    // ---- stage pred-box meta for the 32 rows ----
    if (tid < N_TILE && (n0 + tid) < Nq) {
        float4 pb = reinterpret_cast<const float4*>(pred_boxes)[n0 + tid];
        float hx = 0.5f * pb.z, hy = 0.5f * pb.w;
        sh_pm[tid * 3 + 0] = make_float4(pb.x - hx, pb.y - hy, pb.x + hx, pb.y + hy);
        sh_pm[tid * 3 + 1] = pb;
        sh_pm[tid * 3 + 2] = make_float4(pb.z * pb.w, 0.f, 0.f, 0.f);
    }

    // ---- per-lane target data (reused across all 32 rows) ----
    const bool tvalid = (t < Nt);
    const int  ts     = tvalid ? t : 0;
    float4 tb = reinterpret_cast<const float4*>(tboxes)[ts];
    int lab = labels[ts];
    if ((unsigned)lab >= (unsigned)NCLASS) lab = 0;
    const float thx = 0.5f * tb.z, thy = 0.5f * tb.w;
    const float tx0 = tb.x - thx, ty0 = tb.y - thy;
    const float tx1 = tb.x + thx, ty1 = tb.y + thy;
    const float ta  = tb.z * tb.w;

#if __has_builtin(__builtin_amdgcn_s_wait_asynccnt)
    __builtin_amdgcn_s_wait_asynccnt(0);
#else
    asm volatile("s_wait_asynccnt 0x0" ::: "memory");
#endif
    __syncthreads();

    const int rmax = (Nq - n0 < N_TILE) ? (Nq - n0) : N_TILE;
    const size_t obase = (size_t)n0 * Nt + (size_t)t;
#pragma unroll 4
    for (int r = 0; r < N_TILE; ++r) {
        float4 pc = sh_pm[r * 3 + 0];                 // ds_load_b128 broadcast
        float4 pb = sh_pm[r * 3 + 1];
        float  pa = sh_pm[r * 3 + 2].x;
        float  cls = sh_cc[r * NCLASS + lab];         // per-lane gather

        // L1 on center-format boxes
        float l1 = fabsf(pb.x - tb.x) + fabsf(pb.y - tb.y)
                 + fabsf(pb.z - tb.z) + fabsf(pb.w - tb.w);

        // GIoU
        float iw = fminf(pc.z, tx1) - fmaxf(pc.x, tx0); iw = fmaxf(iw, 0.f);
        float ih = fminf(pc.w, ty1) - fmaxf(pc.y, ty0); ih = fmaxf(ih, 0.f);
        float inter = iw * ih;
        float uni   = pa + ta - inter;
        float iou   = inter * fast_rcp(uni);
        float ew = fmaxf(pc.z, tx1) - fminf(pc.x, tx0); ew = fmaxf(ew, 0.f);
        float eh = fmaxf(pc.w, ty1) - fminf(pc.y, ty0); eh = fmaxf(eh, 0.f);
        float ae = ew * eh;
        float giou = iou - (ae - uni) * fast_rcp(ae);

        float cost = fmaf(BBOX_W, l1, cls) - GIOU_W * giou;
        if (tvalid && r < rmax)
            __builtin_nontemporal_store(cost, &out[obase + (size_t)r * Nt]);
    }
}

// ---------------------------------------------------------------------------
extern "C" void kernel_launch(void* const* d_in, const int* in_sizes, int n_in,
                              void* d_out, int out_size, void* d_ws, size_t ws_size,
                              hipStream_t stream)
{
    (void)n_in; (void)out_size; (void)ws_size;
    const float* logits = (const float*)d_in[0];   // (B,Q,C) f32
    const float* pred   = (const float*)d_in[1];   // (B,Q,4) f32
    const int*   labels = (const int*)d_in[2];     // (T,)    int
    const float* tbox   = (const float*)d_in[3];   // (T,4)   f32
    float* out = (float*)d_out;
    float* cc  = (float*)d_ws;                     // N*NCLASS floats (2.56 MB)

    const int Nq = in_sizes[1] / 4;                // 8000
    const int Nt = in_sizes[3] / 4;                // 3200
    const int total = Nq * NCLASS;

    cc_precompute_kernel<<<(total + 255) / 256, 256, 0, stream>>>(logits, cc, total);

    dim3 grid((Nt + T_TILE - 1) / T_TILE, (Nq + N_TILE - 1) / N_TILE);
    cost_kernel<<<grid, 256, 0, stream>>>(pred, labels, tbox, cc, out, Nq, Nt);
}